// AutoRegressiveGraphConvLayer_62637803045407
// MI455X (gfx1250) — compile-verified
//
#include <hip/hip_runtime.h>
#include <hip/hip_bf16.h>

// ---------------------------------------------------------------------------
// Problem constants (match reference module)
// ---------------------------------------------------------------------------
constexpr int N_   = 1024;
constexpr int M_   = 32;
constexpr int FN_  = 64;
constexpr int FE_  = 32;
constexpr int NE_  = 32240;          // sum_{i=1}^{1023} min(i,32)
constexpr int B_   = 2;
constexpr int TRI_ = (M_ * (M_ + 1)) / 2;   // 528: edges of nodes 1..32
constexpr int ROWS_E = B_ * NE_;     // 64480 (divisible by 32)
constexpr int ROWS_N = B_ * N_;      // 2048  (divisible by 16)

typedef float v2f __attribute__((ext_vector_type(2)));
typedef float v8f __attribute__((ext_vector_type(8)));

// ---------------------------------------------------------------------------
// Closed-form graph indexing (replaces host-built N1/ET/N2/PREV_* tables)
// ---------------------------------------------------------------------------
__device__ __forceinline__ int node_edge_start(int i) {
    return (i <= M_ + 1) ? (i * (i - 1)) / 2 : (TRI_ + (i - (M_ + 1)) * M_);
}

__device__ __forceinline__ void edge_to_ij(int e, int& i, int& j) {
    if (e < TRI_) {
        int ii = (int)((1.0f + sqrtf(8.0f * (float)e + 1.0f)) * 0.5f);
        while (ii * (ii - 1) / 2 > e) --ii;          // guard fp rounding
        while ((ii + 1) * ii / 2 <= e) ++ii;
        i = ii;
        j = e - ii * (ii - 1) / 2;                   // j0 == 0 for i <= 32
    } else {
        int q = (e - TRI_) / M_;
        int r = (e - TRI_) % M_;
        i = (M_ + 1) + q;
        j = (i - M_) + r;
    }
}

// ---------------------------------------------------------------------------
// WMMA helpers: fp32 16x16x4 (full reference precision)
// ---------------------------------------------------------------------------
__device__ __forceinline__ v8f wmma4(v2f a, v2f b, v8f c) {
    return __builtin_amdgcn_wmma_f32_16x16x4_f32(
        /*neg_a=*/false, a, /*neg_b=*/false, b,
        /*c_mod=*/(short)0, c, /*reuse_a=*/false, /*reuse_b=*/false);
}

__device__ __forceinline__ v8f gemm_tile(const float* __restrict__ ldsA, int lda,
                                         const float* __restrict__ W,
                                         int Kd, int Nout, int n0) {
    const int lane = threadIdx.x & 31;
    const int hi   = lane >> 4;      // 0: K pair {0,1}, 1: K pair {2,3}
    const int mn   = lane & 15;      // A row / B col within tile
    v8f acc = {0.f, 0.f, 0.f, 0.f, 0.f, 0.f, 0.f, 0.f};
    for (int k = 0; k < Kd; k += 4) {
        const int ka = k + 2 * hi;
        v2f a, b;
        a.x = ldsA[mn * lda + ka];
        a.y = ldsA[mn * lda + ka + 1];
        b.x = W[(long)ka * Nout + n0 + mn];
        b.y = W[(long)(ka + 1) * Nout + n0 + mn];
        acc = wmma4(a, b, acc);
    }
    return acc;
}

__device__ __forceinline__ void tile_relu_to_lds(v8f acc, const float* __restrict__ bias,
                                                 float* __restrict__ ldsH, int ldh, int n0) {
    const int lane = threadIdx.x & 31;
    const int hi = lane >> 4, n = lane & 15;
    const float bv = bias[n0 + n];
#pragma unroll
    for (int v = 0; v < 8; ++v) {
        float x = acc[v] + bv;
        ldsH[(v + 8 * hi) * ldh + n0 + n] = fmaxf(x, 0.f);
    }
}

__device__ __forceinline__ void tile_relu_to_global(v8f acc, const float* __restrict__ bias,
                                                    float* __restrict__ out, long row0,
                                                    int rowStride, int n0) {
    const int lane = threadIdx.x & 31;
    const int hi = lane >> 4, n = lane & 15;
    const float bv = bias[n0 + n];
#pragma unroll
    for (int v = 0; v < 8; ++v) {
        float x = acc[v] + bv;
        out[(row0 + v + 8 * hi) * (long)rowStride + n0 + n] = fmaxf(x, 0.f);
    }
}

// ---------------------------------------------------------------------------
// CDNA5 async global -> LDS (ASYNCcnt-tracked, no VGPR staging)
// ---------------------------------------------------------------------------
__device__ __forceinline__ void async_copy_b32(const float* gsrc, const float* lds_dst) {
    // low 32 bits of a generic pointer into the LDS aperture == LDS byte offset
    const unsigned lds_off = (unsigned)(size_t)lds_dst;
    asm volatile("global_load_async_to_lds_b32 %0, %1, off"
                 :: "v"(lds_off), "v"(gsrc) : "memory");
}

__device__ __forceinline__ void wait_async_all() {
    asm volatile("s_wait_asynccnt 0x0" ::: "memory");
}

// gather one 16x160 triple tile into LDS via async-to-LDS (all lanes active)
__device__ __forceinline__ void async_gather_triples(float* __restrict__ sbuf, long row0,
                                                     const float* __restrict__ nodes,
                                                     const float* __restrict__ edges) {
    for (int idx = threadIdx.x; idx < 16 * 160; idx += 256) {
        const int r = idx / 160, f = idx % 160;
        const long g = row0 + r;
        const int b = (int)(g / NE_), e = (int)(g % NE_);
        int i, j; edge_to_ij(e, i, j);
        const float* src;
        if (f < FN_)            src = &nodes[((long)b * N_ + j) * FN_ + f];
        else if (f < FN_ + FE_) src = &edges[((long)b * NE_ + e) * FE_ + (f - FN_)];
        else                    src = &nodes[((long)b * N_ + i) * FN_ + (f - FN_ - FE_)];
        async_copy_b32(src, &sbuf[r * 161 + f]);
    }
}

// ---------------------------------------------------------------------------
// Kernel 1: node-branch aggregation MLP over triples [160]->256->128 (hn)
//           2 tiles per block; async gather of tile u+1 overlaps WMMA of tile u
// ---------------------------------------------------------------------------
__global__ __launch_bounds__(256) void k_node_agg(
    const float* __restrict__ nodes, const float* __restrict__ edges,
    const float* __restrict__ W1, const float* __restrict__ b1,
    const float* __restrict__ W2, const float* __restrict__ b2,
    float* __restrict__ hn) {
    __shared__ float sA[2][16 * 161];   // double-buffered A tiles
    __shared__ float sH[16 * 257];      // hidden 16 x 256 (+1 pad)
    const long t0 = (long)blockIdx.x * 2;

    __builtin_prefetch(W1, 0, 3);       // warm L2 ahead of the k-loops
    __builtin_prefetch(W2, 0, 3);

    async_gather_triples(&sA[0][0], t0 * 16, nodes, edges);
    wait_async_all();
    __syncthreads();

#pragma unroll
    for (int u = 0; u < 2; ++u) {
        if (u == 0)   // prefetch next tile while this one computes
            async_gather_triples(&sA[1][0], (t0 + 1) * 16, nodes, edges);

        const float* A = &sA[u][0];
        const long row0 = (t0 + u) * 16;
        const int w = threadIdx.x >> 5;             // 8 waves
        // layer 1: K=160 -> 256 cols (16 tiles, 2 per wave)
        v8f a0 = gemm_tile(A, 161, W1, 160, 256, w * 16);
        v8f a1 = gemm_tile(A, 161, W1, 160, 256, (w + 8) * 16);
        tile_relu_to_lds(a0, b1, sH, 257, w * 16);
        tile_relu_to_lds(a1, b1, sH, 257, (w + 8) * 16);
        __syncthreads();
        // layer 2: K=256 -> 128 cols (8 tiles, 1 per wave)
        v8f a2 = gemm_tile(sH, 257, W2, 256, 128, w * 16);
        tile_relu_to_global(a2, b2, hn, row0, 128, w * 16);

        if (u == 0) {
            wait_async_all();           // tile 1's A landed in LDS
            __syncthreads();            // + protects sH reuse (WAR after reads)
        }
    }
}

// ---------------------------------------------------------------------------
// Kernel 2: windowed node aggregation (contiguous block sum) + concat input
// ---------------------------------------------------------------------------
__global__ __launch_bounds__(192) void k_prev_node(
    const float* __restrict__ hn, const float* __restrict__ nodes,
    float* __restrict__ nodes_in) {
    const int gi = blockIdx.x;               // b*N + i
    const int b = gi / N_, i = gi % N_;
    const int t = (i < M_) ? i : M_;
    const int k0 = node_edge_start(i);
    const float norm = (i >= 1 && i < M_) ? (1.0f / (float)i) : (1.0f / (float)M_);
    const int f = threadIdx.x;
    float* dst = nodes_in + (long)gi * 192;
    if (f < 128) {
        float acc = 0.f;
        const float* src = hn + ((long)b * NE_ + k0) * 128 + f;
        for (int r = 0; r < t; ++r) acc += src[(long)r * 128];
        dst[f] = acc * norm;
    } else {
        dst[f] = nodes[((long)b * N_ + i) * FN_ + (f - 128)];
    }
}

// ---------------------------------------------------------------------------
// Kernel 3: node output MLP [192]->192->64 -> out_nodes
// ---------------------------------------------------------------------------
__global__ __launch_bounds__(256) void k_node_out(
    const float* __restrict__ nodes_in,
    const float* __restrict__ W1, const float* __restrict__ b1,
    const float* __restrict__ W2, const float* __restrict__ b2,
    float* __restrict__ outp) {
    __shared__ float sA[16 * 193];
    __shared__ float sH[16 * 193];
    const long row0 = (long)blockIdx.x * 16;
    for (int idx = threadIdx.x; idx < 16 * 192; idx += 256) {
        const int r = idx / 192, f = idx % 192;
        sA[r * 193 + f] = nodes_in[(row0 + r) * 192 + f];
    }
    __syncthreads();
    const int w = threadIdx.x >> 5;
    v8f accs[2];
    int s = 0;
    for (int ct = w; ct < 12; ct += 8, ++s) accs[s] = gemm_tile(sA, 193, W1, 192, 192, ct * 16);
    s = 0;
    for (int ct = w; ct < 12; ct += 8, ++s) tile_relu_to_lds(accs[s], b1, sH, 193, ct * 16);
    __syncthreads();
    if (w < 4) {
        v8f a2 = gemm_tile(sH, 193, W2, 192, 64, w * 16);
        tile_relu_to_global(a2, b2, outp, row0, 64, w * 16);
    }
}

// ---------------------------------------------------------------------------
// Kernel 4: edge-branch aggregation MLP over pairs [96]->128->64 (he)
// ---------------------------------------------------------------------------
__global__ __launch_bounds__(256) void k_edge_agg(
    const float* __restrict__ nodes, const float* __restrict__ edges,
    const float* __restrict__ W1, const float* __restrict__ b1,
    const float* __restrict__ W2, const float* __restrict__ b2,
    float* __restrict__ he) {
    __shared__ float sA[16 * 97];
    __shared__ float sH[16 * 129];
    const long row0 = (long)blockIdx.x * 16;
    for (int idx = threadIdx.x; idx < 16 * 96; idx += 256) {
        const int r = idx / 96, f = idx % 96;
        const long g = row0 + r;
        const int b = (int)(g / NE_), e = (int)(g % NE_);
        int i, j; edge_to_ij(e, i, j);
        float v;
        if (f < FN_) v = nodes[((long)b * N_ + j) * FN_ + f];
        else         v = edges[((long)b * NE_ + e) * FE_ + (f - FN_)];
        sA[r * 97 + f] = v;
    }
    __syncthreads();
    const int w = threadIdx.x >> 5;
    v8f a0 = gemm_tile(sA, 97, W1, 96, 128, w * 16);     // 8 tiles
    tile_relu_to_lds(a0, b1, sH, 129, w * 16);
    __syncthreads();
    if (w < 4) {
        v8f a2 = gemm_tile(sH, 129, W2, 128, 64, w * 16);
        tile_relu_to_global(a2, b2, he, row0, 64, w * 16);
    }
}

// ---------------------------------------------------------------------------
// Kernel 5: windowed edge aggregation == segmented exclusive prefix scan
// ---------------------------------------------------------------------------
__global__ __launch_bounds__(64) void k_prev_edge(
    const float* __restrict__ he, float* __restrict__ pe) {
    const int gi = blockIdx.x;               // b*N + i
    const int b = gi / N_, i = gi % N_;
    const int t = (i < M_) ? i : M_;
    if (t == 0) return;                      // node 0: no edges
    const int k0 = node_edge_start(i);
    const int f = threadIdx.x;               // 0..63
    float run = 0.f;
    for (int r = 0; r < t; ++r) {
        const long row = (long)b * NE_ + k0 + r;
        const float nrm = (r > 0) ? (1.0f / (float)r) : 1.0f;
        pe[row * 64 + f] = run * nrm;        // exclusive prefix, normalized
        run += he[row * 64 + f];
    }
}

// ---------------------------------------------------------------------------
// Kernel 6: edge output MLP [96]->96->32 -> out_edges
// ---------------------------------------------------------------------------
__global__ __launch_bounds__(256) void k_edge_out(
    const float* __restrict__ pe, const float* __restrict__ edges,
    const float* __restrict__ W1, const float* __restrict__ b1,
    const float* __restrict__ W2, const float* __restrict__ b2,
    float* __restrict__ outp) {
    __shared__ float sA[16 * 97];
    __shared__ float sH[16 * 97];
    const long row0 = (long)blockIdx.x * 16;
    for (int idx = threadIdx.x; idx < 16 * 96; idx += 256) {
        const int r = idx / 96, f = idx % 96;
        const long g = row0 + r;
        float v;
        if (f < 64) v = pe[g * 64 + f];
        else        v = edges[g * FE_ + (f - 64)];   // g == b*NE+e flat
        sA[r * 97 + f] = v;
    }
    __syncthreads();
    const int w = threadIdx.x >> 5;
    if (w < 6) {
        v8f a0 = gemm_tile(sA, 97, W1, 96, 96, w * 16);  // 6 tiles
        tile_relu_to_lds(a0, b1, sH, 97, w * 16);
    }
    __syncthreads();
    if (w < 2) {
        v8f a2 = gemm_tile(sH, 97, W2, 96, 32, w * 16);  // 2 tiles
        tile_relu_to_global(a2, b2, outp, row0, 32, w * 16);
    }
}

// ---------------------------------------------------------------------------
// Launch
// ---------------------------------------------------------------------------
extern "C" void kernel_launch(void* const* d_in, const int* in_sizes, int n_in,
                              void* d_out, int out_size, void* d_ws, size_t ws_size,
                              hipStream_t stream) {
    const float* in_nodes = (const float*)d_in[0];
    const float* in_edges = (const float*)d_in[1];
    const float* Wan1 = (const float*)d_in[2];  const float* ban1 = (const float*)d_in[3];
    const float* Wan2 = (const float*)d_in[4];  const float* ban2 = (const float*)d_in[5];
    const float* Wln1 = (const float*)d_in[6];  const float* bln1 = (const float*)d_in[7];
    const float* Wln2 = (const float*)d_in[8];  const float* bln2 = (const float*)d_in[9];
    const float* Wae1 = (const float*)d_in[10]; const float* bae1 = (const float*)d_in[11];
    const float* Wae2 = (const float*)d_in[12]; const float* bae2 = (const float*)d_in[13];
    const float* Wle1 = (const float*)d_in[14]; const float* ble1 = (const float*)d_in[15];
    const float* Wle2 = (const float*)d_in[16]; const float* ble2 = (const float*)d_in[17];

    float* out_nodes = (float*)d_out;                          // [B,N,64]
    float* out_edges = (float*)d_out + (long)B_ * N_ * 64;     // [B,NE,32]

    // workspace layout (node branch first, then reuse for edge branch)
    float* ws       = (float*)d_ws;
    float* hn       = ws;                              // 64480*128 fp32 (33 MB)
    float* nodes_in = ws + (long)ROWS_E * 128;         // 2048*192
    float* he       = ws;                              // 64480*64 (reuses hn)
    float* pe       = ws + (long)ROWS_E * 64;          // 64480*64

    // --- node branch ---
    k_node_agg<<<ROWS_E / 32, 256, 0, stream>>>(in_nodes, in_edges,
                                                Wan1, ban1, Wan2, ban2, hn);
    k_prev_node<<<ROWS_N, 192, 0, stream>>>(hn, in_nodes, nodes_in);
    k_node_out<<<ROWS_N / 16, 256, 0, stream>>>(nodes_in, Wln1, bln1, Wln2, bln2,
                                                out_nodes);
    // --- edge branch (reuses hn's workspace) ---
    k_edge_agg<<<ROWS_E / 16, 256, 0, stream>>>(in_nodes, in_edges,
                                                Wae1, bae1, Wae2, bae2, he);
    k_prev_edge<<<ROWS_N, 64, 0, stream>>>(he, pe);
    k_edge_out<<<ROWS_E / 16, 256, 0, stream>>>(pe, in_edges,
                                                Wle1, ble1, Wle2, ble2, out_edges);
}